// GeometricLoss_69080253989492
// MI455X (gfx1250) — compile-verified
//
#include <hip/hip_runtime.h>
#include <math.h>

typedef __attribute__((ext_vector_type(2))) float v2f;
typedef __attribute__((ext_vector_type(8))) float v8f;

#define TPB 256
#define NORM_EPS 1e-6f
#define COS_EPS  1e-8f

// Guaranteed native CDNA5 fp32 scatter-add: non-returning GLOBAL_ATOMIC_ADD_F32.
// Fire-and-forget (tracked by STOREcnt; s_endpgm's implicit wait-idle covers it).
__device__ __forceinline__ void atomAddF(float* p, float v) {
    asm volatile("global_atomic_add_f32 %0, %1, off"
                 :
                 : "v"(p), "v"(v)
                 : "memory");
}

// ---------------- K0: zero accumulators, init finite flag ----------------
__global__ void k_zero(float* __restrict__ buf, long long n, unsigned int* flag) {
    long long i = (long long)blockIdx.x * blockDim.x + threadIdx.x;
    long long stride = (long long)gridDim.x * blockDim.x;
    for (; i < n; i += stride) buf[i] = 0.0f;
    if (blockIdx.x == 0 && threadIdx.x == 0) *flag = 1u;
}

// ---------------- K1: face normals scatter (both meshes) ----------------
__global__ void k_face_normals(const float* __restrict__ Vr,
                               const float* __restrict__ Vg,
                               const int*   __restrict__ faces,
                               float* __restrict__ vn_r,
                               float* __restrict__ vn_g,
                               int F) {
    int i = blockIdx.x * blockDim.x + threadIdx.x;
    if (i >= F) return;
    int a = faces[3 * i + 0];
    int b = faces[3 * i + 1];
    int c = faces[3 * i + 2];

    // ground-truth mesh
    {
        float x0 = Vg[3 * a], y0 = Vg[3 * a + 1], z0 = Vg[3 * a + 2];
        float x1 = Vg[3 * b], y1 = Vg[3 * b + 1], z1 = Vg[3 * b + 2];
        float x2 = Vg[3 * c], y2 = Vg[3 * c + 1], z2 = Vg[3 * c + 2];
        float e1x = x1 - x0, e1y = y1 - y0, e1z = z1 - z0;
        float e2x = x2 - x0, e2y = y2 - y0, e2z = z2 - z0;
        float nx = e1y * e2z - e1z * e2y;
        float ny = e1z * e2x - e1x * e2z;
        float nz = e1x * e2y - e1y * e2x;
        atomAddF(&vn_g[3 * a + 0], nx); atomAddF(&vn_g[3 * a + 1], ny); atomAddF(&vn_g[3 * a + 2], nz);
        atomAddF(&vn_g[3 * b + 0], nx); atomAddF(&vn_g[3 * b + 1], ny); atomAddF(&vn_g[3 * b + 2], nz);
        atomAddF(&vn_g[3 * c + 0], nx); atomAddF(&vn_g[3 * c + 1], ny); atomAddF(&vn_g[3 * c + 2], nz);
    }
    // reconstructed mesh
    {
        float x0 = Vr[3 * a], y0 = Vr[3 * a + 1], z0 = Vr[3 * a + 2];
        float x1 = Vr[3 * b], y1 = Vr[3 * b + 1], z1 = Vr[3 * b + 2];
        float x2 = Vr[3 * c], y2 = Vr[3 * c + 1], z2 = Vr[3 * c + 2];
        float e1x = x1 - x0, e1y = y1 - y0, e1z = z1 - z0;
        float e2x = x2 - x0, e2y = y2 - y0, e2z = z2 - z0;
        float nx = e1y * e2z - e1z * e2y;
        float ny = e1z * e2x - e1x * e2z;
        float nz = e1x * e2y - e1y * e2x;
        atomAddF(&vn_r[3 * a + 0], nx); atomAddF(&vn_r[3 * a + 1], ny); atomAddF(&vn_r[3 * a + 2], nz);
        atomAddF(&vn_r[3 * b + 0], nx); atomAddF(&vn_r[3 * b + 1], ny); atomAddF(&vn_r[3 * b + 2], nz);
        atomAddF(&vn_r[3 * c + 0], nx); atomAddF(&vn_r[3 * c + 1], ny); atomAddF(&vn_r[3 * c + 2], nz);
    }
}

// ---------------- K2: Laplacian SpMM scatter (both meshes) ----------------
__global__ void k_spmm(const float* __restrict__ Vr,
                       const float* __restrict__ Vg,
                       const int*   __restrict__ rows,
                       const int*   __restrict__ cols,
                       const float* __restrict__ vals,
                       float* __restrict__ dl_r,
                       float* __restrict__ dl_g,
                       int nnz) {
    int i = blockIdx.x * blockDim.x + threadIdx.x;
    if (i >= nnz) return;
    int r = rows[i];
    int c = cols[i];
    float v = vals[i];
    atomAddF(&dl_g[3 * r + 0], v * Vg[3 * c + 0]);
    atomAddF(&dl_g[3 * r + 1], v * Vg[3 * c + 1]);
    atomAddF(&dl_g[3 * r + 2], v * Vg[3 * c + 2]);
    atomAddF(&dl_r[3 * r + 0], v * Vr[3 * c + 0]);
    atomAddF(&dl_r[3 * r + 1], v * Vr[3 * c + 1]);
    atomAddF(&dl_r[3 * r + 2], v * Vr[3 * c + 2]);
}

__device__ __forceinline__ float cos_sim3(float ax, float ay, float az,
                                          float bx, float by, float bz) {
    float dot = ax * bx + ay * by + az * bz;
    float na = sqrtf(ax * ax + ay * ay + az * az);
    float nb = sqrtf(bx * bx + by * by + bz * bz);
    float cs = dot / (fmaxf(na, COS_EPS) * fmaxf(nb, COS_EPS));
    if (isnan(cs)) cs = 1.0f;   // jnp.nan_to_num(..., nan=1.0)
    return cs;
}

// ---------------- K3: per-vertex loss partials, block tree-reduce ----------------
__global__ void k_vertex(const float* __restrict__ Vr,
                         const float* __restrict__ Vg,
                         const float* __restrict__ vn_r,
                         const float* __restrict__ vn_g,
                         const float* __restrict__ dl_r,
                         const float* __restrict__ dl_g,
                         float* __restrict__ p_l1,
                         float* __restrict__ p_n,
                         float* __restrict__ p_l,
                         unsigned int* __restrict__ finite_flag,
                         int N) {
    __shared__ float s1[TPB], s2[TPB], s3[TPB];
    int tid = threadIdx.x;
    int i = blockIdx.x * TPB + tid;
    float l1 = 0.0f, ln = 0.0f, ll = 0.0f;
    bool fin = true;
    if (i < N) {
        // L1 on positions
        l1 = fabsf(Vr[3 * i + 0] - Vg[3 * i + 0]) +
             fabsf(Vr[3 * i + 1] - Vg[3 * i + 1]) +
             fabsf(Vr[3 * i + 2] - Vg[3 * i + 2]);

        // normalized vertex normals, then cosine
        float gx = vn_g[3 * i], gy = vn_g[3 * i + 1], gz = vn_g[3 * i + 2];
        float rx = vn_r[3 * i], ry = vn_r[3 * i + 1], rz = vn_r[3 * i + 2];
        float ig = 1.0f / fmaxf(sqrtf(gx * gx + gy * gy + gz * gz), NORM_EPS);
        float ir = 1.0f / fmaxf(sqrtf(rx * rx + ry * ry + rz * rz), NORM_EPS);
        gx *= ig; gy *= ig; gz *= ig;
        rx *= ir; ry *= ir; rz *= ir;
        ln = 1.0f - cos_sim3(rx, ry, rz, gx, gy, gz);

        // Laplacian deltas cosine
        float agx = dl_g[3 * i], agy = dl_g[3 * i + 1], agz = dl_g[3 * i + 2];
        float arx = dl_r[3 * i], ary = dl_r[3 * i + 1], arz = dl_r[3 * i + 2];
        ll = 1.0f - cos_sim3(arx, ary, arz, agx, agy, agz);
        fin = isfinite(agx) && isfinite(agy) && isfinite(agz) &&
              isfinite(arx) && isfinite(ary) && isfinite(arz);
    }
    s1[tid] = l1; s2[tid] = ln; s3[tid] = ll;
    __syncthreads();
    for (int off = TPB / 2; off > 0; off >>= 1) {
        if (tid < off) {
            s1[tid] += s1[tid + off];
            s2[tid] += s2[tid + off];
            s3[tid] += s3[tid + off];
        }
        __syncthreads();
    }
    if (tid == 0) {
        p_l1[blockIdx.x] = s1[0];
        p_n[blockIdx.x]  = s2[0];
        p_l[blockIdx.x]  = s3[0];
    }
    if (!fin) atomicAnd(finite_flag, 0u);
}

// WMMA f32 16x16x4 wave-wide reduction: B = ones -> D rows are row-sums of A.
// Each lane contributes a.x (a.y = 0); 32 lane values -> 16 row sums.
// Lanes 0..15 hold rows 0..7 in d[0..7]; lanes 16..31 hold rows 8..15.
__device__ __forceinline__ float wave_reduce_wmma(float s) {
    v2f a; a.x = s;    a.y = 0.0f;
    v2f b; b.x = 1.0f; b.y = 1.0f;
    v8f c = {};
    v8f d = __builtin_amdgcn_wmma_f32_16x16x4_f32(
        /*neg_a=*/false, a, /*neg_b=*/false, b,
        /*c_mod=*/(short)0, c, /*reuse_a=*/false, /*reuse_b=*/false);
    float t = d[0] + d[1] + d[2] + d[3] + d[4] + d[5] + d[6] + d[7];
    return __shfl(t, 0, 32) + __shfl(t, 16, 32);
}

// ---------------- K4: final reduction (single block, 8 waves) ----------------
__global__ void k_final(const float* __restrict__ p_l1,
                        const float* __restrict__ p_n,
                        const float* __restrict__ p_l,
                        const unsigned int* __restrict__ finite_flag,
                        float* __restrict__ out,
                        int nb, int N) {
    __shared__ float acc[3][8];
    int tid = threadIdx.x;
    float s1 = 0.0f, s2 = 0.0f, s3 = 0.0f;
    for (int j = tid; j < nb; j += TPB) {
        s1 += p_l1[j];
        s2 += p_n[j];
        s3 += p_l[j];
    }
    // uniform control flow: every wave executes the WMMAs with EXEC all-ones
    float w1 = wave_reduce_wmma(s1);
    float w2 = wave_reduce_wmma(s2);
    float w3 = wave_reduce_wmma(s3);
    int wave = tid >> 5;
    if ((tid & 31) == 0) {
        acc[0][wave] = w1; acc[1][wave] = w2; acc[2][wave] = w3;
    }
    __syncthreads();
    if (tid == 0) {
        float S1 = 0.0f, S2 = 0.0f, S3 = 0.0f;
        for (int w = 0; w < 8; ++w) { S1 += acc[0][w]; S2 += acc[1][w]; S3 += acc[2][w]; }
        float fN = (float)N;
        float loss_l1 = S1 / (3.0f * fN);
        float loss_n  = S2 / fN;
        float loss_lap = (*finite_flag) ? (S3 / fN) : 0.0f;
        out[0] = 1.0f * loss_l1 + 1.0f * loss_n + 0.1f * loss_lap;
    }
}

extern "C" void kernel_launch(void* const* d_in, const int* in_sizes, int n_in,
                              void* d_out, int out_size, void* d_ws, size_t ws_size,
                              hipStream_t stream) {
    (void)n_in; (void)out_size; (void)ws_size;
    const float* Vr    = (const float*)d_in[0];
    const float* Vg    = (const float*)d_in[1];
    const int*   faces = (const int*)d_in[2];
    const int*   Lr    = (const int*)d_in[3];
    const int*   Lc    = (const int*)d_in[4];
    const float* Lv    = (const float*)d_in[5];

    const int N   = in_sizes[0] / 3;
    const int F   = in_sizes[2] / 3;
    const int NNZ = in_sizes[3];

    float* ws    = (float*)d_ws;
    float* vn_r  = ws;
    float* vn_g  = ws + 3LL * N;
    float* dl_r  = ws + 6LL * N;
    float* dl_g  = ws + 9LL * N;
    const int nb3 = (N + TPB - 1) / TPB;
    float* p_l1  = ws + 12LL * N;
    float* p_n   = p_l1 + nb3;
    float* p_l   = p_n + nb3;
    unsigned int* flag = (unsigned int*)(p_l + nb3);

    const long long zero_n = 12LL * N;
    k_zero<<<4096, TPB, 0, stream>>>(ws, zero_n, flag);
    k_face_normals<<<(F + TPB - 1) / TPB, TPB, 0, stream>>>(Vr, Vg, faces, vn_r, vn_g, F);
    k_spmm<<<(NNZ + TPB - 1) / TPB, TPB, 0, stream>>>(Vr, Vg, Lr, Lc, Lv, dl_r, dl_g, NNZ);
    k_vertex<<<nb3, TPB, 0, stream>>>(Vr, Vg, vn_r, vn_g, dl_r, dl_g,
                                      p_l1, p_n, p_l, flag, N);
    k_final<<<1, TPB, 0, stream>>>(p_l1, p_n, p_l, flag, (float*)d_out, nb3, N);
}